// RecursiveDecoder_62148176773776
// MI455X (gfx1250) — compile-verified
//
#include <hip/hip_runtime.h>
#include <cstddef>
#include <cstdint>

// Dims from reference: F_=256, H=256, C=128, T=4, I=2, S=57, P=10, G=32
#define DF   256
#define DH   256
#define DC   128
#define DT   4
#define DI   2
#define DS_  57
#define DP   10
#define WNE_LD 772   // 3*H + T

typedef float v2f __attribute__((ext_vector_type(2)));
typedef float v8f __attribute__((ext_vector_type(8)));

__device__ __forceinline__ float lrelu(float v) { return v > 0.f ? v : 0.01f * v; }

__device__ __forceinline__ float wave_sum(float s) {
#pragma unroll
  for (int off = 16; off > 0; off >>= 1) s += __shfl_xor(s, off, 32);
  return s;
}

// y[o] = lrelu(dot(v[0:256], W[o,:]) + b[o]); one wave per output row.
// Lane l owns 8 contiguous floats -> two global_load_b128 per lane, 1KB/wave/step.
__global__ void k_matvec_lrelu(const float* __restrict__ v,
                               const float* __restrict__ W,
                               const float* __restrict__ b,
                               float* __restrict__ out, int out_ld) {
  int wave = (blockIdx.x * blockDim.x + threadIdx.x) >> 5;  // 0..32767
  int lane = threadIdx.x & 31;
  int k0 = lane * 8;
  const float4* wv = (const float4*)(W + (size_t)wave * DF + k0);
  const float4* vv = (const float4*)(v + k0);
  float4 w0 = wv[0], w1 = wv[1];
  float4 v0 = vv[0], v1 = vv[1];
  float s = w0.x * v0.x + w0.y * v0.y + w0.z * v0.z + w0.w * v0.w
          + w1.x * v1.x + w1.y * v1.y + w1.z * v1.z + w1.w * v1.w;
  s = wave_sum(s);
  if (lane == 0) {
    float r = lrelu(s + b[wave]);
    out[(size_t)(wave >> 8) * out_ld + (wave & 255)] = r;
  }
}

// Generic fp32 WMMA GEMM: Y[m,n] = act( sum_k X[m, xoff+k]*W[n, woff+k] + bias[n] )
// One 16x16 tile per wave; 4 waves (128 threads) per block along M.
// A tile (16x4): lanes 0-15 hold K={k,k+1} of rows m0+lane; lanes 16-31 K={k+2,k+3}.
// B tile (4x16): lanes 0-15 hold K={k,k+1} of cols n0+lane; lanes 16-31 K={k+2,k+3}.
__global__ void k_gemm_wmma(const float* __restrict__ X, int ldx, int xoff,
                            const float* __restrict__ W, int ldw, int woff,
                            const float* __restrict__ bias,
                            float* __restrict__ Y, int ldy, int K, int act) {
  int waveid = threadIdx.x >> 5;
  int lane   = threadIdx.x & 31;
  int half   = lane >> 4;
  int lmod   = lane & 15;
  int m0 = (blockIdx.y * 4 + waveid) * 16;
  int n0 = blockIdx.x * 16;
  const float* xr = X + (size_t)(m0 + lmod) * ldx + xoff + half * 2;
  const float* wr = W + (size_t)(n0 + lmod) * ldw + woff + half * 2;
  v8f acc = {0.f, 0.f, 0.f, 0.f, 0.f, 0.f, 0.f, 0.f};
  for (int k = 0; k < K; k += 4) {
    v2f a, b;
    a.x = xr[k]; a.y = xr[k + 1];
    b.x = wr[k]; b.y = wr[k + 1];
    acc = __builtin_amdgcn_wmma_f32_16x16x4_f32(false, a, false, b,
                                                (short)0, acc, false, false);
  }
  int n = n0 + lmod;
  float bb = bias ? bias[n] : 0.f;
#pragma unroll
  for (int r = 0; r < 8; ++r) {
    int m = m0 + r + half * 8;
    float val = acc[r] + bb;
    if (act) val = lrelu(val);
    Y[(size_t)m * ldy + n] = val;
  }
}

// E = el @ Wcol.T with el[i*128+j, k] = lrelu(U[i,k]+V[j,k]+bel[k]) built on the fly.
// Wcol[n,k] = Wne_it[n*772 + 512 + k]. M=16384, N=256, K=256.
// B panel (16 rows x 256 cols = 16KB) is staged into LDS once per block with
// CDNA5 async-to-LDS copies (ASYNCcnt), then shared by the block's 4 waves.
__global__ void k_egemm_wmma(const float* __restrict__ U,
                             const float* __restrict__ V,
                             const float* __restrict__ bel,
                             const float* __restrict__ Wit,
                             float* __restrict__ E) {
  __shared__ float smw[16 * 256];  // B panel: smw[r*256 + k] = Wit[n0+r][512+k]
  int tid    = threadIdx.x;
  int waveid = tid >> 5;
  int lane   = tid & 31;
  int half   = lane >> 4;
  int lmod   = lane & 15;
  int m0 = (blockIdx.y * 4 + waveid) * 16;   // 16 | 128 => whole tile shares i
  int n0 = blockIdx.x * 16;

  // Stage B panel: 1024 x 16B chunks, 8 per thread. Row byte stride 772*4=3088
  // (16B aligned) and the 512-float column offset keep every b128 aligned.
#pragma unroll
  for (int rep = 0; rep < 8; ++rep) {
    int chunk = rep * 128 + tid;     // 0..1023
    int r  = chunk >> 6;             // 64 chunks per 256-float row
    int c4 = chunk & 63;             // 16B chunk within row
    const float* src = Wit + (size_t)(n0 + r) * WNE_LD + 512 + c4 * 4;
    uint32_t dst = (uint32_t)(uintptr_t)&smw[r * 256 + c4 * 4];
    uint64_t ga  = (uint64_t)(uintptr_t)src;
    asm volatile("global_load_async_to_lds_b128 %0, %1, off"
                 :: "v"(dst), "v"(ga) : "memory");
  }
  asm volatile("s_wait_asynccnt 0x0" ::: "memory");
  __syncthreads();

  int i = m0 >> 7;
  int j = (m0 & 127) + lmod;
  const float* ur = U + (size_t)i * DH + half * 2;
  const float* vr = V + (size_t)j * DH + half * 2;
  const float* br = bel + half * 2;
  const float* wl = &smw[lmod * 256 + half * 2];
  v8f acc = {0.f, 0.f, 0.f, 0.f, 0.f, 0.f, 0.f, 0.f};
  for (int k = 0; k < DH; k += 4) {
    v2f a, b;
    a.x = lrelu(ur[k]     + vr[k]     + br[k]);
    a.y = lrelu(ur[k + 1] + vr[k + 1] + br[k + 1]);
    b.x = wl[k]; b.y = wl[k + 1];
    acc = __builtin_amdgcn_wmma_f32_16x16x4_f32(false, a, false, b,
                                                (short)0, acc, false, false);
  }
  int n = n0 + lmod;
#pragma unroll
  for (int r = 0; r < 8; ++r) {
    int m = m0 + r + half * 8;
    E[(size_t)m * DH + n] = acc[r];
  }
}

// exists / sem / semins heads: 68 linear outputs per node row, one wave each.
__global__ void k_heads(const float* __restrict__ xbuf,
                        const float* __restrict__ We, const float* __restrict__ be,
                        const float* __restrict__ Ws, const float* __restrict__ bs,
                        const float* __restrict__ Wsi, const float* __restrict__ bsi,
                        float* __restrict__ o_exists, float* __restrict__ o_sem,
                        float* __restrict__ o_semins) {
  int wave = (blockIdx.x * blockDim.x + threadIdx.x) >> 5;
  int lane = threadIdx.x & 31;
  if (wave >= DC * 68) return;
  int row = wave / 68, col = wave % 68;
  const float* wr; float bb; float* outp;
  if (col == 0)      { wr = We;                       bb = be[0];       outp = o_exists + row; }
  else if (col < 58) { int c = col - 1;  wr = Ws  + (size_t)c * DH; bb = bs[c];  outp = o_sem    + (size_t)row * DS_ + c; }
  else               { int c = col - 58; wr = Wsi + (size_t)c * DH; bb = bsi[c]; outp = o_semins + (size_t)row * DP  + c; }
  const float* xr = xbuf + (size_t)row * 768;
  float s = 0.f;
#pragma unroll 2
  for (int k = lane; k < DH; k += 32) s += xr[k] * wr[k];
  s = wave_sum(s);
  if (lane == 0) *outp = s + bb;
}

// edge_logits[i,j,t] = dot(lrelu(U[i]+V[j]+bel), Wee[t]) + bee[t]; one wave per (i,j).
__global__ void k_edges(const float* __restrict__ U, const float* __restrict__ V,
                        const float* __restrict__ bel,
                        const float* __restrict__ Wee, const float* __restrict__ bee,
                        float* __restrict__ elog) {
  int wave = (blockIdx.x * blockDim.x + threadIdx.x) >> 5;  // 0..16383
  int lane = threadIdx.x & 31;
  int i = wave >> 7, j = wave & 127;
  const float* ur = U + (size_t)i * DH;
  const float* vr = V + (size_t)j * DH;
  float s0 = 0.f, s1 = 0.f, s2 = 0.f, s3 = 0.f;
  for (int h = lane; h < DH; h += 32) {
    float e = lrelu(ur[h] + vr[h] + bel[h]);
    s0 += e * Wee[h];
    s1 += e * Wee[DH + h];
    s2 += e * Wee[2 * DH + h];
    s3 += e * Wee[3 * DH + h];
  }
  s0 = wave_sum(s0); s1 = wave_sum(s1); s2 = wave_sum(s2); s3 = wave_sum(s3);
  if (lane == 0) {
    float* o = elog + (size_t)wave * DT;
    o[0] = s0 + bee[0]; o[1] = s1 + bee[1]; o[2] = s2 + bee[2]; o[3] = s3 + bee[3];
  }
}

// group-norm (G=32, 8 ch/group) + affine + lrelu; one row per block.
__global__ void k_groupnorm(const float* __restrict__ gnin,
                            const float* __restrict__ gw, const float* __restrict__ gb,
                            float* __restrict__ out) {
  __shared__ float sm[DF];
  int i = blockIdx.x, h = threadIdx.x;
  float v = gnin[(size_t)i * DF + h];
  sm[h] = v;
  __syncthreads();
  int g0 = h & ~7;
  float mean = 0.f, m2 = 0.f;
#pragma unroll
  for (int k = 0; k < 8; ++k) { float x = sm[g0 + k]; mean += x; m2 += x * x; }
  mean *= 0.125f;
  m2 = m2 * 0.125f - mean * mean;
  float xn = (v - mean) * rsqrtf(m2 + 1e-5f);
  out[(size_t)i * DF + h] = lrelu(xn * gw[h] + gb[h]);
}

// x_new[i,h] = max(0, max_{j,t in mask} lrelu(A[i,h]+B[j,h]+E[i,j,h]+logit*Wt[h,t]+bne[h]))
__global__ void k_mpreduce(const float* __restrict__ Ab, const float* __restrict__ Bb,
                           const float* __restrict__ E,
                           const float* __restrict__ bne_it, const float* __restrict__ Wit,
                           const float* __restrict__ exists, const float* __restrict__ elog,
                           float* __restrict__ xout) {
  int i = blockIdx.x, h = threadIdx.x;
  float m = 0.f;
  if (exists[i] > 0.f) {
    float a = Ab[(size_t)i * DH + h] + bne_it[h];
    const float* wt = Wit + (size_t)h * WNE_LD + 768;
    float w0 = wt[0], w1 = wt[1], w2 = wt[2], w3 = wt[3];
    for (int j = 0; j < DC; ++j) {
      if (exists[j] <= 0.f) continue;               // wave-uniform branch
      float base = a + Bb[(size_t)j * DH + h] + E[((size_t)(i * DC + j)) * DH + h];
      const float* lg = elog + (size_t)(i * DC + j) * DT;
      float l0 = lg[0], l1 = lg[1], l2 = lg[2], l3 = lg[3];
      if (l0 > 0.f) m = fmaxf(m, lrelu(base + l0 * w0));
      if (l1 > 0.f) m = fmaxf(m, lrelu(base + l1 * w1));
      if (l2 > 0.f) m = fmaxf(m, lrelu(base + l2 * w2));
      if (l3 > 0.f) m = fmaxf(m, lrelu(base + l3 * w3));
    }
  }
  xout[(size_t)i * 768 + h] = m;
}

extern "C" void kernel_launch(void* const* d_in, const int* in_sizes, int n_in,
                              void* d_out, int out_size, void* d_ws, size_t ws_size,
                              hipStream_t stream) {
  (void)in_sizes; (void)n_in; (void)out_size; (void)ws_size;
  const float* psf = (const float*)d_in[0];
  const float* pgf = (const float*)d_in[1];
  const float* Wp  = (const float*)d_in[2];
  const float* bp  = (const float*)d_in[3];
  const float* We  = (const float*)d_in[4];
  const float* be  = (const float*)d_in[5];
  const float* Ws  = (const float*)d_in[6];
  const float* bs  = (const float*)d_in[7];
  const float* Wsi = (const float*)d_in[8];
  const float* bsi = (const float*)d_in[9];
  const float* Wel = (const float*)d_in[10];
  const float* bel = (const float*)d_in[11];
  const float* Wee = (const float*)d_in[12];
  const float* bee = (const float*)d_in[13];
  const float* Wne = (const float*)d_in[14];
  const float* bne = (const float*)d_in[15];
  const float* Wc  = (const float*)d_in[16];
  const float* bc  = (const float*)d_in[17];
  const float* Wc2 = (const float*)d_in[18];
  const float* bc2 = (const float*)d_in[19];
  const float* Wgc = (const float*)d_in[20];
  const float* bgc = (const float*)d_in[21];
  const float* Wgp = (const float*)d_in[22];
  const float* bgp = (const float*)d_in[23];
  const float* gw  = (const float*)d_in[24];
  const float* gb  = (const float*)d_in[25];

  float* out       = (float*)d_out;
  float* o_child   = out;            // 128*256
  float* o_geo     = out + 32768;    // 128*256
  float* o_sem     = out + 65536;    // 128*57
  float* o_semins  = out + 72832;    // 128*10
  float* o_exists  = out + 74112;    // 128
  float* o_elog    = out + 74240;    // 128*128*4

  float* ws   = (float*)d_ws;
  float* xbuf = ws;            // 128 x 768 (feats[0..2] concatenated per row)
  float* U    = ws + 98304;    // 128 x 256
  float* V    = ws + 131072;   // 128 x 256
  float* pg   = ws + 163840;   // 128 x 256
  float* gnin = ws + 196608;   // 128 x 256
  float* Ab   = ws + 229376;   // 128 x 256
  float* Bb   = ws + 262144;   // 128 x 256
  float* ch   = ws + 294912;   // 128 x 256
  float* E    = ws + 327680;   // 16384 x 256  (~16.8 MB; total ws ~18.1 MB)

  // 1) cf = lrelu(psf @ Wp.T + bp) -> xbuf[:, 0:256]   (HBM-bound: 33.5 MB weights)
  k_matvec_lrelu<<<4096, 256, 0, stream>>>(psf, Wp, bp, xbuf, 768);
  // 2) pg = lrelu(pgf @ Wgp.T + bgp)
  k_matvec_lrelu<<<4096, 256, 0, stream>>>(pgf, Wgp, bgp, pg, 256);
  // 3) gnin = pg @ Wgc.T + bgc   (WMMA)
  k_gemm_wmma<<<dim3(16, 2), 128, 0, stream>>>(pg, 256, 0, Wgc, 256, 0, bgc,
                                               gnin, 256, 256, 0);
  // 4) child_geo = lrelu(groupnorm(gnin))
  k_groupnorm<<<128, 256, 0, stream>>>(gnin, gw, gb, o_geo);
  // 5) exists / sem / semins heads
  k_heads<<<1088, 256, 0, stream>>>(xbuf, We, be, Ws, bs, Wsi, bsi,
                                    o_exists, o_sem, o_semins);
  // 6) U = cf @ Wel[:, :256].T ; V = cf @ Wel[:, 256:].T   (WMMA)
  k_gemm_wmma<<<dim3(16, 2), 128, 0, stream>>>(xbuf, 768, 0, Wel, 512, 0, nullptr,
                                               U, 256, 256, 0);
  k_gemm_wmma<<<dim3(16, 2), 128, 0, stream>>>(xbuf, 768, 0, Wel, 512, 256, nullptr,
                                               V, 256, 256, 0);
  // 7) edge_logits from on-the-fly el = lrelu(U_i + V_j + bel)
  k_edges<<<2048, 256, 0, stream>>>(U, V, bel, Wee, bee, o_elog);
  // 8) message-passing iterations
  for (int it = 0; it < DI; ++it) {
    const float* Wit = Wne + (size_t)it * DH * WNE_LD;
    const float* bit = bne + (size_t)it * DH;
    // A = x @ Wne[:, :256].T ; B = x @ Wne[:, 256:512].T   (WMMA)
    k_gemm_wmma<<<dim3(16, 2), 128, 0, stream>>>(xbuf, 768, it * 256, Wit, WNE_LD, 0,
                                                 nullptr, Ab, 256, 256, 0);
    k_gemm_wmma<<<dim3(16, 2), 128, 0, stream>>>(xbuf, 768, it * 256, Wit, WNE_LD, 256,
                                                 nullptr, Bb, 256, 256, 0);
    // E = el @ Wne[:, 512:768].T  (dominant 2.1-GFLOP WMMA GEMM, async-staged B)
    k_egemm_wmma<<<dim3(16, 256), 128, 0, stream>>>(U, V, bel, Wit, E);
    // masked scatter-max -> feats[it+1]
    k_mpreduce<<<128, 256, 0, stream>>>(Ab, Bb, E, bit, Wit, o_exists, o_elog,
                                        xbuf + (size_t)(it + 1) * 256);
  }
  // 9) ch = lrelu(cat(feats) @ Wc.T + bc)   (K=768, WMMA)
  k_gemm_wmma<<<dim3(16, 2), 128, 0, stream>>>(xbuf, 768, 0, Wc, 768, 0, bc,
                                               ch, 256, 768, 1);
  // 10) child = lrelu(ch @ Wc2.T + bc2)
  k_gemm_wmma<<<dim3(16, 2), 128, 0, stream>>>(ch, 256, 0, Wc2, 256, 0, bc2,
                                               o_child, 256, 256, 1);
}